// NonOverlappingLocallyConnected1d_24885040513342
// MI455X (gfx1250) — compile-verified
//
#include <hip/hip_runtime.h>

// ---- problem constants (match reference) ----
#define BS      16
#define CIN     32
#define COUT    64
#define OUT_DIM 1024
#define KPATCH  4
#define DLEN    (OUT_DIM * KPATCH)   // 4096
#define SCALE   0.17677669529663687f // 1/sqrt(32)

typedef __attribute__((ext_vector_type(2))) float v2f;
typedef __attribute__((ext_vector_type(8))) float v8f;

// One wave per patch p: full 16(b) x 64(o) tile, K=128 via 32 x
// v_wmma_f32_16x16x4_f32 per 16x16 sub-tile (4 sub-tiles over cout).
// Each K=4 WMMA slice == one channel c's 4 contiguous floats -> float2/lane.
__global__ __launch_bounds__(128) void
nolc1d_wmma_f32(const float* __restrict__ x,
                const float* __restrict__ w,
                float* __restrict__ out)
{
    const int lane = threadIdx.x & 31;
    const int wvid = threadIdx.x >> 5;          // 4 waves / block
    const int p    = (blockIdx.x << 2) + wvid;  // 256 blocks * 4 = 1024 patches

    const int half = lane >> 4;    // 0 -> K={0,1}, 1 -> K={2,3} (ISA 32-bit A/B layout)
    const int lid  = lane & 15;    // A: M=batch row, B: N=cout col, C/D: N=cout col

    // x[b][c][4p+kk] ; b = lid            (A fragment, 16x4 f32)
    const float* xp = x + ((size_t)lid * CIN) * DLEN + 4 * p + 2 * half;
    // w[o][c][4p+kk] ; o = t*16 + lid     (B fragment = w^T, 4x16 f32)
    const float* wp = w + ((size_t)lid * CIN) * DLEN + 4 * p + 2 * half;
    const size_t tstride = (size_t)16 * CIN * DLEN; // next cout tile (16 rows of w)

    v8f acc0 = {}, acc1 = {}, acc2 = {}, acc3 = {};

#pragma unroll 4
    for (int c = 0; c < CIN; ++c) {
        const size_t co = (size_t)c * DLEN;
        v2f a  = *(const v2f*)(xp + co);
        v2f b0 = *(const v2f*)(wp + co);
        v2f b1 = *(const v2f*)(wp + co + tstride);
        v2f b2 = *(const v2f*)(wp + co + 2 * tstride);
        v2f b3 = *(const v2f*)(wp + co + 3 * tstride);

        // 8-arg pattern: (neg_a, A, neg_b, B, c_mod, C, reuse_a, reuse_b)
        acc0 = __builtin_amdgcn_wmma_f32_16x16x4_f32(false, a, false, b0,
                                                     (short)0, acc0, false, false);
        acc1 = __builtin_amdgcn_wmma_f32_16x16x4_f32(false, a, false, b1,
                                                     (short)0, acc1, false, false);
        acc2 = __builtin_amdgcn_wmma_f32_16x16x4_f32(false, a, false, b2,
                                                     (short)0, acc2, false, false);
        acc3 = __builtin_amdgcn_wmma_f32_16x16x4_f32(false, a, false, b3,
                                                     (short)0, acc3, false, false);
    }

    // C/D layout: VGPR j, lanes 0-15 -> M=j ; lanes 16-31 -> M=j+8 ; N=lid.
    // out[b][o][p], flat idx = (b*COUT + o)*OUT_DIM + p
    const int b_base = half * 8;
#pragma unroll
    for (int j = 0; j < 8; ++j) {
        const int b = b_base + j;
        const size_t row = (size_t)b * COUT * OUT_DIM + p;
        out[row + (size_t)(0 * 16 + lid) * OUT_DIM] = acc0[j] * SCALE;
        out[row + (size_t)(1 * 16 + lid) * OUT_DIM] = acc1[j] * SCALE;
        out[row + (size_t)(2 * 16 + lid) * OUT_DIM] = acc2[j] * SCALE;
        out[row + (size_t)(3 * 16 + lid) * OUT_DIM] = acc3[j] * SCALE;
    }
}

extern "C" void kernel_launch(void* const* d_in, const int* in_sizes, int n_in,
                              void* d_out, int out_size, void* d_ws, size_t ws_size,
                              hipStream_t stream)
{
    (void)in_sizes; (void)n_in; (void)out_size; (void)d_ws; (void)ws_size;
    const float* x = (const float*)d_in[0];   // [16, 32, 4096] fp32
    const float* w = (const float*)d_in[1];   // [64, 32, 4096] fp32
    float* out     = (float*)d_out;           // [16, 64, 1024] fp32

    dim3 grid(OUT_DIM / 4);   // 256 blocks
    dim3 block(128);          // 4 waves, one patch p per wave
    nolc1d_wmma_f32<<<grid, block, 0, stream>>>(x, w, out);
}